// GCN_12154757447857
// MI455X (gfx1250) — compile-verified
//
#include <hip/hip_runtime.h>
#include <hip/hip_bf16.h>

typedef float v2f __attribute__((ext_vector_type(2)));
typedef float v8f __attribute__((ext_vector_type(8)));

#define BN_EPS 1e-5f

// ---------------- degree / symmetric norm ----------------
__global__ void k_deg_init(float* __restrict__ deg, int n) {
  int i = blockIdx.x * blockDim.x + threadIdx.x;
  if (i < n) deg[i] = 1.0f;  // self-loop contribution
}

__global__ void k_deg_edges(const long long* __restrict__ dst, float* __restrict__ deg,
                            long long e) {
  long long i = (long long)blockIdx.x * blockDim.x + threadIdx.x;
  if (i < e) atomicAdd(&deg[dst[i]], 1.0f);
}

__global__ void k_dinv(float* __restrict__ d, int n) {
  int i = blockIdx.x * blockDim.x + threadIdx.x;
  if (i < n) d[i] = rsqrtf(d[i]);  // deg >= 1 always (self-loops)
}

// ---------------- WMMA f32 GEMM: C[M,NOUT] = A[M,K] @ B[K,NOUT] ----------------
// blockDim.x = 32*(NOUT/16), gridDim.x = M/16 (M multiple of 16; K multiple of 4).
// One wave owns one 16x16 C tile; V_WMMA_F32_16X16X4_F32 stepped over K.
// Layout per ISA 7.12.2:
//   A 16x4 : lanes 0-15 -> (K=kk+0,kk+1), lanes 16-31 -> (K=kk+2,kk+3), M = lane&15
//   B 4x16 : vgpr0 -> K=kk+(half*2), vgpr1 -> K=kk+(half*2)+1, N = lane&15
//   C      : vgpr i -> M = i + half*8, N = lane&15
__global__ void k_gemm_wmma_f32(const float* __restrict__ A, const float* __restrict__ B,
                                const float* __restrict__ bias, float* __restrict__ C,
                                int M, int K, int NOUT, int fuse_bias_relu) {
  const int wave = threadIdx.x >> 5;
  const int lane = threadIdx.x & 31;
  const int half = lane >> 4;
  const int l15  = lane & 15;
  const int r0 = blockIdx.x << 4;
  const int c0 = wave << 4;

  v8f acc = {};
  const float* ap = A + (size_t)(r0 + l15) * K + (half << 1);
  const float* bp = B + (size_t)(half << 1) * NOUT + c0 + l15;
  for (int kk = 0; kk < K; kk += 4) {
    v2f a, b;
    a.x = ap[0];
    a.y = ap[1];
    b.x = bp[0];
    b.y = bp[NOUT];
    ap += 4;
    bp += (size_t)4 * NOUT;
    acc = __builtin_amdgcn_wmma_f32_16x16x4_f32(false, a, false, b, (short)0, acc,
                                                false, false);
  }

  float bv = fuse_bias_relu ? bias[c0 + l15] : 0.0f;
  float* cp = C + (size_t)(r0 + (half << 3)) * NOUT + c0 + l15;
#pragma unroll
  for (int i = 0; i < 8; ++i) {
    float v = acc[i] + bv;
    if (fuse_bias_relu) v = fmaxf(v, 0.0f);
    cp[(size_t)i * NOUT] = v;
  }
}

// ---------------- self-loop init: agg = hw * dinv[node]^2 (fully writes agg) ----------------
__global__ void k_self_init(const float* __restrict__ hw, const float* __restrict__ dinv,
                            float* __restrict__ agg, int fshift, long long total) {
  long long i = (long long)blockIdx.x * blockDim.x + threadIdx.x;
  if (i >= total) return;
  float dn = dinv[i >> fshift];
  agg[i] = hw[i] * dn * dn;
}

// ---------------- edge scatter: one wave per edge, FPL floats per lane ----------------
template <int FPL>  // feature width F = 32*FPL (FPL=4 -> 128, FPL=2 -> 64)
__global__ void k_edge_agg(const long long* __restrict__ src, const long long* __restrict__ dst,
                           const float* __restrict__ hw, const float* __restrict__ dinv,
                           float* __restrict__ agg, long long e) {
  long long w = ((long long)blockIdx.x * blockDim.x + threadIdx.x) >> 5;
  int lane = threadIdx.x & 31;
  if (w >= e) return;
  long long s = src[w];
  long long d = dst[w];
  float nrm = dinv[s] * dinv[d];
  const int shift = (FPL == 4) ? 7 : 6;
  const float* hp = hw + ((size_t)s << shift) + lane * FPL;   // coalesced 512B/wave
  float* ap       = agg + ((size_t)d << shift) + lane * FPL;
  float vals[FPL];
#pragma unroll
  for (int i = 0; i < FPL; ++i) vals[i] = hp[i];
#pragma unroll
  for (int i = 0; i < FPL; ++i) atomicAdd(&ap[i], vals[i] * nrm);  // global_atomic_add_f32
}

// ---------------- fused bias + eval-BN + ReLU (in place) ----------------
__global__ void k_bias_bn_relu(float* __restrict__ x, const float* __restrict__ bias,
                               const float* __restrict__ g, const float* __restrict__ beta,
                               const float* __restrict__ m, const float* __restrict__ v,
                               int fmask, long long total) {
  long long i = (long long)blockIdx.x * blockDim.x + threadIdx.x;
  if (i >= total) return;
  int f = (int)(i & fmask);
  float val = x[i] + bias[f];
  val = (val - m[f]) * rsqrtf(v[f] + BN_EPS) * g[f] + beta[f];
  x[i] = fmaxf(val, 0.0f);
}

// ---------------- head: logits = h @ w_out + b_out ; log_softmax ----------------
__global__ void k_head(const float* __restrict__ h, const float* __restrict__ w,
                       const float* __restrict__ b, float* __restrict__ out, int n) {
  __shared__ float ws[64 * 8];
  __shared__ float bs[8];
  for (int i = threadIdx.x; i < 64 * 8; i += blockDim.x) ws[i] = w[i];
  if (threadIdx.x < 8) bs[threadIdx.x] = b[threadIdx.x];
  __syncthreads();

  int r = blockIdx.x * blockDim.x + threadIdx.x;
  if (r >= n) return;
  float acc[8];
#pragma unroll
  for (int c = 0; c < 8; ++c) acc[c] = bs[c];
  const float* hp = h + (size_t)r * 64;
  for (int k = 0; k < 64; ++k) {
    float hv = hp[k];
#pragma unroll
    for (int c = 0; c < 8; ++c) acc[c] += hv * ws[k * 8 + c];
  }
  float mx = acc[0];
#pragma unroll
  for (int c = 1; c < 8; ++c) mx = fmaxf(mx, acc[c]);
  float se = 0.0f;
#pragma unroll
  for (int c = 0; c < 8; ++c) se += expf(acc[c] - mx);
  float lse = logf(se) + mx;
  float* op = out + (size_t)r * 8;
#pragma unroll
  for (int c = 0; c < 8; ++c) op[c] = acc[c] - lse;
}

extern "C" void kernel_launch(void* const* d_in, const int* in_sizes, int n_in,
                              void* d_out, int out_size, void* d_ws, size_t ws_size,
                              hipStream_t stream) {
  (void)n_in; (void)out_size; (void)ws_size;
  // setup_inputs() dict order
  const float*     x    = (const float*)d_in[0];
  const long long* ei   = (const long long*)d_in[1];  // int64 [2,E]
  const float* w_in  = (const float*)d_in[2];
  const float* b_in  = (const float*)d_in[3];
  const float* w1    = (const float*)d_in[4];
  const float* b1    = (const float*)d_in[5];
  const float* w2    = (const float*)d_in[6];
  const float* b2    = (const float*)d_in[7];
  const float* w3    = (const float*)d_in[8];
  const float* b3    = (const float*)d_in[9];
  const float* w_out = (const float*)d_in[10];
  const float* b_out = (const float*)d_in[11];
  const float* g1  = (const float*)d_in[12];
  const float* be1 = (const float*)d_in[13];
  const float* m1  = (const float*)d_in[14];
  const float* v1  = (const float*)d_in[15];
  const float* g2  = (const float*)d_in[16];
  const float* be2 = (const float*)d_in[17];
  const float* m2  = (const float*)d_in[18];
  const float* v2  = (const float*)d_in[19];
  const float* g3  = (const float*)d_in[20];
  const float* be3 = (const float*)d_in[21];
  const float* m3  = (const float*)d_in[22];
  const float* v3  = (const float*)d_in[23];
  float* out = (float*)d_out;

  const int N = in_sizes[0] / 128;                  // 100000, multiple of 16
  const long long E = (long long)in_sizes[1] / 2;   // 1600000
  const long long* src = ei;
  const long long* dst = ei + E;

  // workspace: dinv[131072] | bufA[N*128] | bufB[N*128]  (~103 MB)
  float* W    = (float*)d_ws;
  float* dinv = W;
  float* bufA = W + (1 << 17);
  float* bufB = bufA + (size_t)N * 128;

  const int t = 256;
  const int nb = (N + t - 1) / t;
  const int eb_thr = (int)((E + t - 1) / t);
  const int eb_wav = (int)((E * 32 + t - 1) / t);   // wave-per-edge launches
  const long long tot128 = (long long)N * 128;
  const long long tot64  = (long long)N * 64;
  const int b128 = (int)((tot128 + t - 1) / t);
  const int b64  = (int)((tot64 + t - 1) / t);

  // degrees & symmetric norm
  k_deg_init<<<nb, t, 0, stream>>>(dinv, N);
  k_deg_edges<<<eb_thr, t, 0, stream>>>(dst, dinv, E);
  k_dinv<<<nb, t, 0, stream>>>(dinv, N);

  // input layer: h0 = relu(x @ w_in + b_in)   [N,128]x[128,64] -> bufA
  k_gemm_wmma_f32<<<N / 16, 32 * (64 / 16), 0, stream>>>(x, w_in, b_in, bufA, N, 128, 64, 1);

  // conv1: 64 -> 128
  k_gemm_wmma_f32<<<N / 16, 32 * (128 / 16), 0, stream>>>(bufA, w1, nullptr, bufB, N, 64, 128, 0);
  k_self_init<<<b128, t, 0, stream>>>(bufB, dinv, bufA, 7, tot128);
  k_edge_agg<4><<<eb_wav, t, 0, stream>>>(src, dst, bufB, dinv, bufA, E);
  k_bias_bn_relu<<<b128, t, 0, stream>>>(bufA, b1, g1, be1, m1, v1, 127, tot128);

  // conv2: 128 -> 128
  k_gemm_wmma_f32<<<N / 16, 32 * (128 / 16), 0, stream>>>(bufA, w2, nullptr, bufB, N, 128, 128, 0);
  k_self_init<<<b128, t, 0, stream>>>(bufB, dinv, bufA, 7, tot128);
  k_edge_agg<4><<<eb_wav, t, 0, stream>>>(src, dst, bufB, dinv, bufA, E);
  k_bias_bn_relu<<<b128, t, 0, stream>>>(bufA, b2, g2, be2, m2, v2, 127, tot128);

  // conv3: 128 -> 64
  k_gemm_wmma_f32<<<N / 16, 32 * (64 / 16), 0, stream>>>(bufA, w3, nullptr, bufB, N, 128, 64, 0);
  k_self_init<<<b64, t, 0, stream>>>(bufB, dinv, bufA, 6, tot64);
  k_edge_agg<2><<<eb_wav, t, 0, stream>>>(src, dst, bufB, dinv, bufA, E);
  k_bias_bn_relu<<<b64, t, 0, stream>>>(bufA, b3, g3, be3, m3, v3, 63, tot64);

  // head: logits + log_softmax -> out
  k_head<<<nb, t, 0, stream>>>(bufA, w_out, b_out, out, N);
}